// MinibatchDiscrimination_37108517437727
// MI455X (gfx1250) — compile-verified
//
#include <hip/hip_runtime.h>
#include <hip/hip_bf16.h>
#include <math.h>

// ---------------------------------------------------------------------------
// MinibatchDiscrimination: out[i] = concat(x_i[0:512], L1[i,0:1024])
//   L1[i,j] = sum_d |x_i[d] - x_j[d]|,  N=1024, D=512, fp32.
// VALU-bound (pairwise L1 has no exact GEMM form -> no WMMA for the math).
// CDNA5 path: async global->LDS (ASYNCcnt) double-buffered tile pipeline.
// Symmetry: L1[i,j]==L1[j,i] -> only upper-triangular 64x64 blocks computed,
// each tile written in both orientations (~1.9x less VALU work).
// ---------------------------------------------------------------------------

#define MBD_N     1024
#define MBD_D     512
#define MBD_OUTW  (MBD_D + MBD_N)   // 1536

#define BM 64
#define BN 64
#define BK 32
#define NCHUNK (MBD_D / BK)         // 16
#define QP (BK / 4)                 // 8 k-quad planes per chunk
#define PLANE (4 * BM + 4)          // 260 dwords per plane (pad 4 -> bank-conflict-free)
#define BUFSZ (QP * PLANE)          // 2080 floats per buffer

// gfx1250 async-load builtin is typed as (global v4i*, local v4i*, Ii, Ii).
typedef int mbd_v4i __attribute__((ext_vector_type(4)));
typedef __attribute__((address_space(1))) mbd_v4i mbd_gv4i;
typedef __attribute__((address_space(3))) mbd_v4i mbd_lv4i;

// Async 16-byte global -> LDS copy (tracked with ASYNCcnt).
__device__ __forceinline__ void mbd_async_copy16(const float* g, float* l) {
#if __has_builtin(__builtin_amdgcn_global_load_async_to_lds_b128)
  __builtin_amdgcn_global_load_async_to_lds_b128(
      (mbd_gv4i*)(void*)g, (mbd_lv4i*)(void*)l, /*imm offset*/ 0, /*cpol*/ 0);
#else
  *(float4*)l = *(const float4*)g;
#endif
}

#if __has_builtin(__builtin_amdgcn_s_wait_asynccnt)
#define MBD_WAIT_ASYNC(n) __builtin_amdgcn_s_wait_asynccnt(n)
#else
#define MBD_WAIT_ASYNC(n) asm volatile("s_wait_asynccnt %0" ::"n"(n) : "memory")
#endif

// ---------------------------------------------------------------------------
// Kernel 1: copy X (N x D) into out[:, 0:D] (row stride OUTW), float4-wide.
// ---------------------------------------------------------------------------
__global__ __launch_bounds__(256) void mbd_copy_kernel(
    const float* __restrict__ x, float* __restrict__ out) {
  const int v = blockIdx.x * 256 + threadIdx.x;       // float4 index, < N*D/4
  const int row = v >> 7;                             // D/4 = 128 float4 per row
  const int c4  = v & 127;
  const float4 val = ((const float4*)x)[v];
  ((float4*)(out + (size_t)row * MBD_OUTW))[c4] = val;
}

// ---------------------------------------------------------------------------
// Kernel 2: 64x64 L1-distance tile per workgroup (upper triangle only),
// async double-buffered LDS, mirrored writeout.
// Thread (tx,ty), tx=tid&15, ty=tid>>4, owns rows {ty+16*mi}, cols {tx+16*ni}.
// LDS layout: element (m,k) at dword (k>>2)*PLANE + 4*m + (k&3).
//   - fill:   lane loads 4 consecutive k (16B, coalesced 128B per 8 lanes),
//             writes 16B contiguous to LDS.
//   - A-frag: ds_load_b128, address depends only on ty -> broadcast.
//   - B-frag: ds_load_b128, banks 4(q+tx)+j -> all 64 banks, conflict-free.
// ---------------------------------------------------------------------------
__global__ __launch_bounds__(256) void mbd_l1_kernel(
    const float* __restrict__ x, float* __restrict__ out) {
  // Upper-triangular blocks only (by <= bx); L1 is symmetric.
  if (blockIdx.y > blockIdx.x) return;

  __shared__ __align__(16) float As[2][BUFSZ];
  __shared__ __align__(16) float Bs[2][BUFSZ];

  const int tid = threadIdx.x;
  const int tx = tid & 15;
  const int ty = tid >> 4;
  const int i0 = blockIdx.y * BM;   // tile rows
  const int j0 = blockIdx.x * BN;   // tile cols (>= i0)

  // Issue the 4 async B128 fills this thread owns for chunk c into buffer buf.
  auto issue = [&](int buf, int c) {
#pragma unroll
    for (int i = 0; i < 2; ++i) {
      const int qi = tid + 256 * i;  // 0..511 quad id
      const int m  = qi >> 3;        // tile row 0..63
      const int q  = qi & 7;         // k-quad 0..7
      const float* ga = x + (size_t)(i0 + m) * MBD_D + c * BK + 4 * q;
      const float* gb = x + (size_t)(j0 + m) * MBD_D + c * BK + 4 * q;
      mbd_async_copy16(ga, &As[buf][q * PLANE + 4 * m]);
      mbd_async_copy16(gb, &Bs[buf][q * PLANE + 4 * m]);
    }
  };

  float acc[4][4];
#pragma unroll
  for (int mi = 0; mi < 4; ++mi)
#pragma unroll
    for (int ni = 0; ni < 4; ++ni) acc[mi][ni] = 0.0f;

  issue(0, 0);

  for (int c = 0; c < NCHUNK; ++c) {
    if (c + 1 < NCHUNK) {
      issue((c + 1) & 1, c + 1);
      // Per-wave in-order completion: <=4 outstanding => chunk c's 4 are done.
      MBD_WAIT_ASYNC(4);
    } else {
      MBD_WAIT_ASYNC(0);
    }
    __syncthreads();

    const float* A = As[c & 1];
    const float* B = Bs[c & 1];

#pragma unroll
    for (int q = 0; q < QP; ++q) {
      float4 a[4], b[4];
#pragma unroll
      for (int mi = 0; mi < 4; ++mi)
        a[mi] = *(const float4*)(A + q * PLANE + 4 * (ty + 16 * mi));
#pragma unroll
      for (int ni = 0; ni < 4; ++ni)
        b[ni] = *(const float4*)(B + q * PLANE + 4 * (tx + 16 * ni));
#pragma unroll
      for (int mi = 0; mi < 4; ++mi)
#pragma unroll
        for (int ni = 0; ni < 4; ++ni) {
          acc[mi][ni] += __builtin_fabsf(a[mi].x - b[ni].x);
          acc[mi][ni] += __builtin_fabsf(a[mi].y - b[ni].y);
          acc[mi][ni] += __builtin_fabsf(a[mi].z - b[ni].z);
          acc[mi][ni] += __builtin_fabsf(a[mi].w - b[ni].w);
        }
    }
    __syncthreads();  // protect buffer (c&1) before it is refilled next iter
  }

  // Epilogue: L1 block -> out[:, D:], plus mirrored block (symmetry).
  // Direct: lanes tx=0..15 write 64B contiguous per (mi,ni).
  // Mirror: scalar scattered stores; total output BW is tiny (4 MB @ 23 TB/s).
  // On diagonal blocks the mirror writes duplicate identical values: benign.
#pragma unroll
  for (int mi = 0; mi < 4; ++mi) {
    const int row = i0 + ty + 16 * mi;
    float* orow = out + (size_t)row * MBD_OUTW + MBD_D + j0;
#pragma unroll
    for (int ni = 0; ni < 4; ++ni) {
      const int col = j0 + tx + 16 * ni;
      const float v = acc[mi][ni];
      orow[tx + 16 * ni] = v;                                  // L1[row][col]
      out[(size_t)col * MBD_OUTW + MBD_D + row] = v;           // L1[col][row]
    }
  }
}

extern "C" void kernel_launch(void* const* d_in, const int* in_sizes, int n_in,
                              void* d_out, int out_size, void* d_ws, size_t ws_size,
                              hipStream_t stream) {
  const float* x = (const float*)d_in[0];
  float* out = (float*)d_out;

  // Columns 0..D-1: passthrough copy of the features.
  mbd_copy_kernel<<<dim3((MBD_N * MBD_D / 4) / 256), dim3(256), 0, stream>>>(x, out);

  // Columns D..D+N-1: all-pairs L1 distances, upper-triangular 64x64 tiles.
  mbd_l1_kernel<<<dim3(MBD_N / BN, MBD_N / BM), dim3(256), 0, stream>>>(x, out);
}